// ParityLSTM_86053964742927
// MI455X (gfx1250) — compile-verified
//
#include <hip/hip_runtime.h>
#include <cstdint>

typedef __attribute__((ext_vector_type(2))) float v2f;
typedef __attribute__((ext_vector_type(8))) float v8f;

#define B_  2048
#define T_  2048
#define I_  9
#define H_  16
#define KPAD 12      // I padded up to multiple of 4 (3 K-chunks of 4)
#define TILE 16

__device__ __forceinline__ float sigm(float x) {
    return 1.0f / (1.0f + __expf(-x));
}

__launch_bounds__(32)
__global__ void ParityLSTM_kernel(const float* __restrict__ x,
                                  const int*   __restrict__ x_lens,
                                  const float* __restrict__ W_ih,   // [64,9]
                                  const float* __restrict__ W_hh,   // [64,16]
                                  const float* __restrict__ b_ih,   // [64]
                                  const float* __restrict__ b_hh,   // [64]
                                  const float* __restrict__ W_lin,  // [2,16]
                                  const float* __restrict__ b_lin,  // [2]
                                  float* __restrict__ out)          // [B,2]
{
    __shared__ float xbuf[2][TILE][13];  // 12 cols padded to 13 (bank spread)
    __shared__ float hbuf[TILE][17];     // 16 cols padded to 17 (bank spread)

    const int lane = threadIdx.x;        // wave32
    const int ln   = lane & 15;          // N index (column) / A-row
    const int hsel = lane >> 4;          // 0: K+=0, 1: K+=2 (WMMA f32 A/B layout)
    const int base = blockIdx.x * TILE;  // first batch row of this tile

    // ---- Load B operands (weights) into registers, WMMA 16x16x4 B layout:
    //      VGPR j, lane-half hsel -> K = 4*kc + 2*hsel + j, N = ln
    float bih[4][3][2];   // W_ih^T chunks (K padded 9->12)
    float bhh[4][4][2];   // W_hh^T chunks (K = 16)
    for (int n = 0; n < 4; ++n) {
        for (int kc = 0; kc < 4; ++kc) {
            for (int j = 0; j < 2; ++j) {
                const int col = 4 * kc + 2 * hsel + j;
                bhh[n][kc][j] = W_hh[(n * 16 + ln) * H_ + col];
                if (kc < 3)
                    bih[n][kc][j] = (col < I_) ? W_ih[(n * 16 + ln) * I_ + col] : 0.0f;
            }
        }
    }
    float bias[4];
    for (int n = 0; n < 4; ++n) bias[n] = b_ih[n * 16 + ln] + b_hh[n * 16 + ln];

    // ---- Sequence lengths for the 8 batch rows this lane owns (D-layout M = g + 8*hsel)
    int lenm1[8];
    int mymax = 0;
    for (int g = 0; g < 8; ++g) {
        lenm1[g] = x_lens[base + g + 8 * hsel] - 1;
        mymax = (lenm1[g] > mymax) ? lenm1[g] : mymax;
    }
    for (int off = 16; off >= 1; off >>= 1) {
        int o = __shfl_xor(mymax, off, 32);
        mymax = (o > mymax) ? o : mymax;
    }
    const int tmax = mymax;

    // ---- Per-slot x staging: 16 rows x 12 cols = 192 slots, 6 per lane
    const float* xp[6];
    int rr[6], cc[6];
    for (int k = 0; k < 6; ++k) {
        const int s = lane + 32 * k;
        rr[k] = s / KPAD;
        cc[k] = s % KPAD;
        xp[k] = x + ((size_t)(base + rr[k]) * T_) * I_ + cc[k];
    }
    // stage t = 0
    for (int k = 0; k < 6; ++k)
        xbuf[0][rr[k]][cc[k]] = (cc[k] < I_) ? xp[k][0] : 0.0f;
    // h(-1) = 0
    for (int g = 0; g < 8; ++g) hbuf[g + 8 * hsel][ln] = 0.0f;

    float cst[8], hfin[8];
    for (int g = 0; g < 8; ++g) { cst[g] = 0.0f; hfin[g] = 0.0f; }

    __syncthreads();  // single-wave block: lowers to cheap sync, keeps LDS ordered

    for (int t = 0; t <= tmax; ++t) {
        const int cur = t & 1, nxt = cur ^ 1;

        // ---- prefetch x tile for t+1 into registers; deep prefetch t+9 into L2
        float pf[6];
        const bool more = (t < tmax);
        for (int k = 0; k < 6; ++k) {
            const float* p = xp[k] + (size_t)(t + 1) * I_;
            pf[k] = (more && cc[k] < I_) ? p[0] : 0.0f;
            __builtin_prefetch(p + 8 * I_, 0, 1);   // -> global_prefetch_b8
        }

        // ---- A operands from LDS (A layout: row M = ln, K = 4*kc + 2*hsel + j)
        v2f ax[3], ah[4];
        for (int kc = 0; kc < 3; ++kc) {
            ax[kc].x = xbuf[cur][ln][4 * kc + 2 * hsel + 0];
            ax[kc].y = xbuf[cur][ln][4 * kc + 2 * hsel + 1];
        }
        for (int kc = 0; kc < 4; ++kc) {
            ah[kc].x = hbuf[ln][4 * kc + 2 * hsel + 0];
            ah[kc].y = hbuf[ln][4 * kc + 2 * hsel + 1];
        }

        // ---- gates = bias + x_t @ W_ih^T + h_{t-1} @ W_hh^T   (4 gate tiles of 16)
        v8f acc[4];
        for (int n = 0; n < 4; ++n) {
            v8f a = { bias[n], bias[n], bias[n], bias[n],
                      bias[n], bias[n], bias[n], bias[n] };
            for (int kc = 0; kc < 3; ++kc) {
                v2f bm = { bih[n][kc][0], bih[n][kc][1] };
                a = __builtin_amdgcn_wmma_f32_16x16x4_f32(
                        false, ax[kc], false, bm, (short)0, a, false, false);
            }
            for (int kc = 0; kc < 4; ++kc) {
                v2f bm = { bhh[n][kc][0], bhh[n][kc][1] };
                a = __builtin_amdgcn_wmma_f32_16x16x4_f32(
                        false, ah[kc], false, bm, (short)0, a, false, false);
            }
            acc[n] = a;
        }

        // ---- commit prefetched x tile to the other LDS buffer
        for (int k = 0; k < 6; ++k)
            xbuf[nxt][rr[k]][cc[k]] = pf[k];

        // ---- elementwise LSTM cell (PyTorch gate order i,f,g,o), D layout
        float hnew[8];
        for (int g = 0; g < 8; ++g) {
            const float iv = sigm(acc[0][g]);
            const float fv = sigm(acc[1][g]);
            const float gv = tanhf(acc[2][g]);
            const float ov = sigm(acc[3][g]);
            const float cv = fv * cst[g] + iv * gv;
            cst[g] = cv;
            const float hv = ov * tanhf(cv);
            hnew[g] = hv;
            hfin[g] = (t == lenm1[g]) ? hv : hfin[g];  // gather h at len-1
        }

        // ---- publish h_t for next step's A operands (D-layout -> LDS)
        for (int g = 0; g < 8; ++g) hbuf[g + 8 * hsel][ln] = hnew[g];
    }

    // ---- final linear head: out[b] = h_final[b] @ W_lin^T + b_lin
    __syncthreads();
    for (int g = 0; g < 8; ++g) hbuf[g + 8 * hsel][ln] = hfin[g];
    __syncthreads();
    if (lane < 16) {
        const int row = lane;
        float o0 = b_lin[0], o1 = b_lin[1];
        for (int n = 0; n < H_; ++n) {
            const float hv = hbuf[row][n];
            o0 += hv * W_lin[n];         // W_lin[0][n]
            o1 += hv * W_lin[H_ + n];    // W_lin[1][n]
        }
        out[(size_t)(base + row) * 2 + 0] = o0;
        out[(size_t)(base + row) * 2 + 1] = o1;
    }
}

extern "C" void kernel_launch(void* const* d_in, const int* in_sizes, int n_in,
                              void* d_out, int out_size, void* d_ws, size_t ws_size,
                              hipStream_t stream) {
    (void)in_sizes; (void)n_in; (void)d_ws; (void)ws_size; (void)out_size;
    const float* x      = (const float*)d_in[0];
    const int*   x_lens = (const int*)  d_in[1];
    const float* W_ih   = (const float*)d_in[2];
    const float* W_hh   = (const float*)d_in[3];
    const float* b_ih   = (const float*)d_in[4];
    const float* b_hh   = (const float*)d_in[5];
    const float* W_lin  = (const float*)d_in[6];
    const float* b_lin  = (const float*)d_in[7];
    float* out = (float*)d_out;

    dim3 grid(B_ / TILE);   // 128 independent batch tiles
    dim3 block(32);         // one wave32 per tile
    ParityLSTM_kernel<<<grid, block, 0, stream>>>(x, x_lens, W_ih, W_hh,
                                                  b_ih, b_hh, W_lin, b_lin, out);
}